// NRBS_7799660610073
// MI455X (gfx1250) — compile-verified
//
#include <hip/hip_runtime.h>

typedef __attribute__((ext_vector_type(2))) float v2f;
typedef __attribute__((ext_vector_type(8))) float v8f;

#define MU_F 1.0f

// ---------------- K1: encoded partials via V_WMMA_F32_16X16X4_F32 ----------------
// encoded = enc_w(32 x N) @ x(N).  Each wave owns (rowblock r in {0,1}, k-chunk c).
// A tile = enc_w[r*16:+16, k:k+4]  (documented f32 16x4 layout:
//   lanes 0-15: V0=K0,V1=K1 ; lanes 16-31: V0=K2,V1=K3 ; M = lane&15)
// B tile = x[k:k+4] broadcast to all 16 columns (same K mapping as A).
// Only D column 0 is consumed: lane0 -> M=0..7 (acc[0..7]), lane16 -> M=8..15.
// Wave-uniform bounds are readfirstlane-scalarized so loop control is SALU and
// EXEC is structurally all-1s at every WMMA (ISA 7.12 requirement).
__global__ void k1_gemv_wmma(const float* __restrict__ enc_w,
                             const float* __restrict__ x,
                             float* __restrict__ wsEnc,
                             int N, int chunkLen) {
  const int lane = threadIdx.x & 31;
  const int wv   = threadIdx.x >> 5;
  const int g    = __builtin_amdgcn_readfirstlane(blockIdx.x * (blockDim.x >> 5) + wv);
  const int r    = g & 1;                                 // row block (0..1)
  const int c    = g >> 1;                                // k-chunk   (0..255)
  const int m    = r * 16 + (lane & 15);
  const int hi   = lane >> 4;
  const int kbeg = __builtin_amdgcn_readfirstlane(c * chunkLen);
  const int kend = __builtin_amdgcn_readfirstlane(min(kbeg + chunkLen, N));
  const int kmain = kbeg + ((kend - kbeg) & ~3);          // full 4-wide iterations

  const float* __restrict__ arow = enc_w + (size_t)m * N;

  v8f acc = {};
  int k = kbeg;
  for (; k < kmain; k += 4) {                             // steady state: no guards
    int ka = k + 2 * hi;
    v2f av = { arow[ka], arow[ka + 1] };
    v2f bv = { x[ka],    x[ka + 1]    };
    acc = __builtin_amdgcn_wmma_f32_16x16x4_f32(false, av, false, bv,
                                                (short)0, acc, false, false);
  }
  if (k < kend) {                                         // wave-uniform tail
    int ka = k + 2 * hi;
    float a0 = 0.f, a1 = 0.f, b0 = 0.f, b1 = 0.f;
    if (ka < kend)     { a0 = arow[ka];     b0 = x[ka];     }
    if (ka + 1 < kend) { a1 = arow[ka + 1]; b1 = x[ka + 1]; }
    v2f av = {a0, a1};
    v2f bv = {b0, b1};
    acc = __builtin_amdgcn_wmma_f32_16x16x4_f32(false, av, false, bv,
                                                (short)0, acc, false, false);
  }
  if (lane == 0 || lane == 16) {
    int base = c * 32 + r * 16 + ((lane == 16) ? 8 : 0);
    #pragma unroll
    for (int j = 0; j < 8; ++j) wsEnc[base + j] = acc[j];
  }
}

// ---------------- K2: reduce encoded partials, bandwidths, inv(w*MU)^2 ----------------
__global__ void k2_reduce_bw(const float* __restrict__ wsEnc, int chunks,
                             const float* __restrict__ enc_b,
                             const float* __restrict__ bw_w,
                             const float* __restrict__ bw_b,
                             float* __restrict__ encoded,
                             float* __restrict__ invw2T) {
  __shared__ float enc[32];
  const int t = threadIdx.x;               // 0..511
  if (t < 32) {
    float s = enc_b[t];
    for (int cc = 0; cc < chunks; ++cc) s += wsEnc[cc * 32 + t];
    enc[t] = s;
    encoded[t] = s;
  }
  __syncthreads();
  float s = bw_b[t];
  #pragma unroll
  for (int j = 0; j < 32; ++j) s += bw_w[t * 32 + j] * enc[j];
  const float wmu = s * MU_F;
  const int band = t >> 4;                 // bandwidths.reshape(32,16): row-major
  const int lab  = t & 15;
  invw2T[lab * 32 + band] = 1.0f / (wmu * wmu);
}

// ---------------- K3: decoder (32 x N) -> dec_t (N x 32), 128B rows ----------------
__global__ void k3_transpose(const float* __restrict__ dec,
                             float* __restrict__ dec_t, int N) {
  __shared__ float tile[32][33];
  const int p0 = blockIdx.x * 32;
  const int tx = threadIdx.x;              // 0..31
  const int ty = threadIdx.y;              // 0..7
  for (int b = ty; b < 32; b += 8) {
    int p = p0 + tx;
    tile[b][tx] = (p < N) ? dec[b * N + p] : 0.f;
  }
  __syncthreads();
  for (int rr = ty; rr < 32; rr += 8) {
    int p = p0 + rr;
    if (p < N) dec_t[p * 32 + tx] = tile[tx][rr];
  }
}

// ---------------- K4: denominator partials: denom[b][k] = sum_p win[b,p,k] ----------------
__global__ void k4_denom(const float* __restrict__ nd,
                         const int* __restrict__ labels,
                         const float* __restrict__ invw2T,
                         float* __restrict__ partials, int N, int totalWaves) {
  __shared__ float red[8][1024];
  const int lane = threadIdx.x & 31;       // lane = band b
  const int wv   = threadIdx.x >> 5;
  const int gw   = blockIdx.x * 8 + wv;
  float acc[32];
  #pragma unroll
  for (int k = 0; k < 32; ++k) acc[k] = 0.f;

  for (int p = gw; p < N; p += totalWaves) {
    float d  = nd[p * 32 + lane];
    float d2 = d * d;
    int lab  = labels[p];
    float iw2 = invw2T[lab * 32 + lane];
    #pragma unroll
    for (int k = 0; k < 32; ++k) {
      float d2k = __shfl(d2, k, 32);
      acc[k] += fmaxf(0.f, 1.f - d2k * iw2);
    }
  }
  #pragma unroll
  for (int k = 0; k < 32; ++k) red[wv][lane * 32 + k] = acc[k];
  __syncthreads();
  for (int j = threadIdx.x; j < 1024; j += 256) {
    float s = 0.f;
    #pragma unroll
    for (int w = 0; w < 8; ++w) s += red[w][j];
    partials[blockIdx.x * 1024 + j] = s;     // j = b*32 + k
  }
}

// ---------------- K5: reduce denom partials -> rdT[k*32 + b] = 1/denom[b][k] ----------------
__global__ void k5_rdenom(const float* __restrict__ partials, int nb,
                          float* __restrict__ rdT) {
  for (int j = threadIdx.x; j < 1024; j += blockDim.x) {
    float s = 0.f;
    for (int c = 0; c < nb; ++c) s += partials[c * 1024 + j];
    int b = j >> 5, k = j & 31;
    rdT[k * 32 + b] = 1.0f / s;
  }
}

// ---------------- K6: output. wave per point, lane = band, coalesced L2 gathers ----------------
template <bool TRANSPOSED>
__global__ void k6_output(const float* __restrict__ nd,
                          const int* __restrict__ ids,
                          const int* __restrict__ labels,
                          const float* __restrict__ invw2T,
                          const float* __restrict__ rdT,
                          const float* __restrict__ encoded,
                          const float* __restrict__ decp,  // dec_t (Nx32) or decoder (32xN)
                          float* __restrict__ out, int N, int totalWaves) {
  const int lane = threadIdx.x & 31;
  const int gw   = blockIdx.x * (blockDim.x >> 5) + (threadIdx.x >> 5);
  const float enc_l = encoded[lane];

  for (int p = gw; p < N; p += totalWaves) {
    float d   = nd[p * 32 + lane];
    float d2  = d * d;
    int   idv = ids[p * 32 + lane];
    int   lab = labels[p];
    float iw2 = invw2T[lab * 32 + lane];
    float acc = 0.f;
    #pragma unroll
    for (int k = 0; k < 32; ++k) {
      float d2k = __shfl(d2, k, 32);
      int   idk = __shfl(idv, k, 32);
      float win = fmaxf(0.f, 1.f - d2k * iw2) * rdT[k * 32 + lane];
      float dv  = TRANSPOSED ? decp[idk * 32 + lane]   // 128B coalesced, L2-resident
                             : decp[lane * N + idk];
      acc += win * dv;
    }
    float v = enc_l * acc;
    #pragma unroll
    for (int off = 16; off > 0; off >>= 1) v += __shfl_xor(v, off, 32);
    if (lane == 0) out[p] = v;
  }
}

extern "C" void kernel_launch(void* const* d_in, const int* in_sizes, int n_in,
                              void* d_out, int out_size, void* d_ws, size_t ws_size,
                              hipStream_t stream) {
  const float* x       = (const float*)d_in[0];
  const float* enc_w   = (const float*)d_in[1];
  const float* enc_b   = (const float*)d_in[2];
  const float* decoder = (const float*)d_in[3];
  const float* bw_w    = (const float*)d_in[4];
  const float* bw_b    = (const float*)d_in[5];
  const float* nd      = (const float*)d_in[6];
  const int*   nid     = (const int*)d_in[7];
  const int*   labels  = (const int*)d_in[8];
  float* out = (float*)d_out;
  const int N = in_sizes[0];

  // ---- workspace layout (floats) ----
  float* ws = (float*)d_ws;
  const int CHUNKS = 256;                        // K1: 128 blocks * 4 waves / 2 rowblocks
  const int PB     = 240;                        // K4 blocks
  size_t offEnc     = 0;                         // CHUNKS*32
  size_t offEncoded = offEnc + (size_t)CHUNKS * 32;
  size_t offInv     = offEncoded + 32;
  size_t offRdT     = offInv + 512;
  size_t offDen     = offRdT + 1024;
  size_t offDecT    = offDen + (size_t)PB * 1024;
  size_t needFast   = (offDecT + (size_t)N * 32) * sizeof(float);
  const bool fast = (ws_size >= needFast);

  float* wsEnc    = ws + offEnc;
  float* encoded  = ws + offEncoded;
  float* invw2T   = ws + offInv;
  float* rdT      = ws + offRdT;
  float* denPart  = ws + offDen;
  float* dec_t    = ws + offDecT;

  int chunkLen = ((N + CHUNKS - 1) / CHUNKS + 3) & ~3;

  if (fast) {
    k3_transpose<<<(N + 31) / 32, dim3(32, 8), 0, stream>>>(decoder, dec_t, N);
  }
  k1_gemv_wmma<<<128, 128, 0, stream>>>(enc_w, x, wsEnc, N, chunkLen);
  k2_reduce_bw<<<1, 512, 0, stream>>>(wsEnc, CHUNKS, enc_b, bw_w, bw_b, encoded, invw2T);
  k4_denom<<<PB, 256, 0, stream>>>(nd, labels, invw2T, denPart, N, PB * 8);
  k5_rdenom<<<1, 256, 0, stream>>>(denPart, PB, rdT);
  if (fast) {
    k6_output<true><<<480, 256, 0, stream>>>(nd, nid, labels, invw2T, rdT, encoded,
                                             dec_t, out, N, 480 * 8);
  } else {
    k6_output<false><<<480, 256, 0, stream>>>(nd, nid, labels, invw2T, rdT, encoded,
                                              decoder, out, N, 480 * 8);
  }
}